// MemNet_60120952209564
// MI455X (gfx1250) — compile-verified
//
#include <hip/hip_runtime.h>
#include <math.h>

typedef __attribute__((ext_vector_type(2))) float v2f;
typedef __attribute__((ext_vector_type(4))) float f4;
typedef __attribute__((ext_vector_type(8))) float v8f;

#define IMG_FEAT 2048
#define HIDDEN   512
#define FUSION   2560

#if __has_builtin(__builtin_amdgcn_global_load_async_to_lds_b128) && \
    __has_builtin(__builtin_amdgcn_s_wait_asynccnt)
#define USE_ASYNC_LDS 1
#else
#define USE_ASYNC_LDS 0
#endif

#if USE_ASYNC_LDS
typedef int v4i_ __attribute__((vector_size(16)));
typedef __attribute__((address_space(1))) v4i_* as1_v4i;
typedef __attribute__((address_space(3))) v4i_* as3_v4i;
#endif

__device__ __forceinline__ void copy16_g2l(const float* g, float* l) {
#if USE_ASYNC_LDS
  __builtin_amdgcn_global_load_async_to_lds_b128(
      (as1_v4i)(void*)g, (as3_v4i)(void*)l, 0, 0);
#else
  *(f4*)l = *(const f4*)g;
#endif
}

__device__ __forceinline__ void async_wait0() {
#if USE_ASYNC_LDS
  __builtin_amdgcn_s_wait_asynccnt(0);
#endif
}

__device__ __forceinline__ float fast_tanh(float x) {
  // tanh(x) = (e^{2x}-1)/(e^{2x}+1); clamp to keep exp finite.
  x = fminf(fmaxf(x, -15.f), 15.f);
  float e = __expf(2.f * x);
  return (e - 1.f) / (e + 1.f);
}

// out[m][n] = [resid[m][n] +] tanh( sum_k X[m][k]*W[n][k] + bias[n] )
// CONCAT: X = [img | ques] along k (2048 + 512). Tile 128(M) x 64(N) per
// 256-thread workgroup; K staged through double-buffered LDS chunks of 32
// via async global->LDS copies (ASYNCcnt) when available.
template <bool CONCAT, bool RESID>
__global__ __launch_bounds__(256) void gemm_tanh_kernel(
    const float* __restrict__ Ximg,   // CONCAT: img [M][2048], else X [M][K]
    const float* __restrict__ Xques,  // CONCAT: ques [M][512], else unused
    const float* __restrict__ W,      // [N][K] (row n = weights over k)
    const float* __restrict__ bias,   // [N]
    const float* __restrict__ resid,  // RESID: [M][N]
    float* __restrict__ out,          // [M][N]
    int M, int N, int K) {
  __shared__ float As[2][128 * 36];  // 128 rows x 32 k, padded stride 36
  __shared__ float Bs[2][64 * 36];   // 64 cols  x 32 k, padded stride 36

  const int t    = threadIdx.x;
  const int wave = t >> 5;
  const int lane = t & 31;
  const int l15  = lane & 15;
  const int hi   = lane >> 4;   // selects K-pair (A/B) and M+8 (C/D)

  const int bm = blockIdx.x * 128;
  const int bn = blockIdx.y * 64;

  const int m0 = (wave >> 1) * 32;  // wave's M offset inside tile
  const int n0 = (wave & 1) * 32;   // wave's N offset inside tile

  v8f zero = {};
  v8f acc[2][2];
#pragma unroll
  for (int mi = 0; mi < 2; ++mi)
#pragma unroll
    for (int ni = 0; ni < 2; ++ni) acc[mi][ni] = zero;

  const int lrow = t >> 3;        // 0..31
  const int lcol = (t & 7) * 4;   // 0,4,...,28

  // Stage one 32-wide K chunk into LDS buffer `buf`.
  auto stage = [&](int buf, int k0) {
    const float* ga[4];
    const float* gb[2];
#pragma unroll
    for (int p = 0; p < 4; ++p) {
      int gm = bm + p * 32 + lrow;
      if (CONCAT) {
        if (k0 < IMG_FEAT)
          ga[p] = Ximg + (long)gm * IMG_FEAT + (k0 + lcol);
        else
          ga[p] = Xques + (long)gm * HIDDEN + (k0 - IMG_FEAT + lcol);
      } else {
        ga[p] = Ximg + (long)gm * K + (k0 + lcol);
      }
    }
#pragma unroll
    for (int p = 0; p < 2; ++p) {
      int gn = bn + p * 32 + lrow;
      gb[p] = W + (long)gn * K + (k0 + lcol);
    }
#if USE_ASYNC_LDS
#pragma unroll
    for (int p = 0; p < 4; ++p)
      copy16_g2l(ga[p], &As[buf][(p * 32 + lrow) * 36 + lcol]);
#pragma unroll
    for (int p = 0; p < 2; ++p)
      copy16_g2l(gb[p], &Bs[buf][(p * 32 + lrow) * 36 + lcol]);
#else
    f4 va[4], vb[2];
#pragma unroll
    for (int p = 0; p < 4; ++p) va[p] = *(const f4*)ga[p];
#pragma unroll
    for (int p = 0; p < 2; ++p) vb[p] = *(const f4*)gb[p];
#pragma unroll
    for (int p = 0; p < 4; ++p)
      *(f4*)(&As[buf][(p * 32 + lrow) * 36 + lcol]) = va[p];
#pragma unroll
    for (int p = 0; p < 2; ++p)
      *(f4*)(&Bs[buf][(p * 32 + lrow) * 36 + lcol]) = vb[p];
#endif
  };

  const int nch = K / 32;
  stage(0, 0);

  for (int ch = 0; ch < nch; ++ch) {
    const int cur = ch & 1;
    async_wait0();        // my async copies into As/Bs[cur] have landed
    __syncthreads();      // everyone's landed; buf[1-cur] no longer being read
    if (ch + 1 < nch) stage(1 - cur, (ch + 1) * 32);

    const float* as = As[cur];
    const float* bs = Bs[cur];
#pragma unroll
    for (int kk = 0; kk < 32; kk += 4) {
      v2f a[2], b[2];
#pragma unroll
      for (int mi = 0; mi < 2; ++mi)
        a[mi] = *(const v2f*)(as + (m0 + mi * 16 + l15) * 36 + kk + 2 * hi);
#pragma unroll
      for (int ni = 0; ni < 2; ++ni)
        b[ni] = *(const v2f*)(bs + (n0 + ni * 16 + l15) * 36 + kk + 2 * hi);
#pragma unroll
      for (int mi = 0; mi < 2; ++mi)
#pragma unroll
        for (int ni = 0; ni < 2; ++ni)
          acc[mi][ni] = __builtin_amdgcn_wmma_f32_16x16x4_f32(
              false, a[mi], false, b[ni], (short)0, acc[mi][ni], false, false);
    }
    __syncthreads();  // done reading buf[cur] before it is restaged
  }

  // ---- epilogue: bias + tanh (+ residual), C/D layout: VGPR i -> row i+8*hi
#pragma unroll
  for (int mi = 0; mi < 2; ++mi) {
#pragma unroll
    for (int ni = 0; ni < 2; ++ni) {
      int gn   = bn + n0 + ni * 16 + l15;
      float bv = bias[gn];
#pragma unroll
      for (int i = 0; i < 8; ++i) {
        int gm  = bm + m0 + mi * 16 + i + 8 * hi;
        float v = fast_tanh(acc[mi][ni][i] + bv);
        if (RESID) v += resid[(long)gm * N + gn];
        out[(long)gm * N + gn] = v;
      }
    }
  }
}

// One workgroup per row b (BR = 5120 blocks). hist[b] staged in LDS once.
// scores[r] = sum_d hist[b][r][d]*fused[b][d]*w_att[d]; softmax over r
// (b_att cancels in softmax); he[b][d] = sum_r attn[r]*hist[b][r][d].
__global__ __launch_bounds__(256) void attn_kernel(
    const float* __restrict__ hist,   // [BR][10][512]
    const float* __restrict__ fused,  // [BR][512]
    const float* __restrict__ w_att,  // [512]
    float* __restrict__ he) {         // [BR][512]
  __shared__ float h[10 * 512];
  __shared__ float fw[512];
  __shared__ float sc[16];

  const int b = blockIdx.x;
  const int t = threadIdx.x;
  const long hbase = (long)b * (10 * 512);

  // stage hist[b] (5120 floats): 5 async 16B copies per thread
#pragma unroll
  for (int i = 0; i < 5; ++i) {
    int idx = (i * 256 + t) * 4;
    copy16_g2l(hist + hbase + idx, h + idx);
  }
  // fw[d] = fused[b][d] * w_att[d]
#pragma unroll
  for (int i = 0; i < 2; ++i) {
    int d = i * 256 + t;
    fw[d] = fused[(long)b * 512 + d] * w_att[d];
  }
  if (t < 16) sc[t] = 0.f;
  async_wait0();
  __syncthreads();

  // per-thread partial scores over d = t and d = t+256
  float s[10];
#pragma unroll
  for (int r = 0; r < 10; ++r)
    s[r] = h[r * 512 + t] * fw[t] + h[r * 512 + t + 256] * fw[t + 256];

  // wave32 reduce + LDS float atomic combine across the 8 waves
#pragma unroll
  for (int r = 0; r < 10; ++r) {
    float v = s[r];
#pragma unroll
    for (int off = 16; off > 0; off >>= 1) v += __shfl_xor(v, off, 32);
    if ((t & 31) == 0) atomicAdd(&sc[r], v);
  }
  __syncthreads();

  // softmax over 10 scores (redundant in every thread)
  float mx = sc[0];
#pragma unroll
  for (int r = 1; r < 10; ++r) mx = fmaxf(mx, sc[r]);
  float e[10];
  float sum = 0.f;
#pragma unroll
  for (int r = 0; r < 10; ++r) {
    e[r] = __expf(sc[r] - mx);
    sum += e[r];
  }
  float inv = 1.f / sum;

  // weighted sum of history rows
#pragma unroll
  for (int i = 0; i < 2; ++i) {
    int d = i * 256 + t;
    float a = 0.f;
#pragma unroll
    for (int r = 0; r < 10; ++r) a += e[r] * h[r * 512 + d];
    he[(long)b * 512 + d] = a * inv;
  }
}

extern "C" void kernel_launch(void* const* d_in, const int* in_sizes, int n_in,
                              void* d_out, int out_size, void* d_ws,
                              size_t ws_size, hipStream_t stream) {
  const float* img    = (const float*)d_in[0];
  const float* ques   = (const float*)d_in[1];
  const float* hist   = (const float*)d_in[2];
  const float* W_fuse = (const float*)d_in[3];
  const float* b_fuse = (const float*)d_in[4];
  const float* w_att  = (const float*)d_in[5];
  // d_in[6] = b_att: constant across softmax axis -> cancels, unused.
  const float* W_hist = (const float*)d_in[7];
  const float* b_hist = (const float*)d_in[8];

  const int B = 512, R = 10, BR = B * R, D = HIDDEN;

  float* fused = (float*)d_ws;              // [BR][D]
  float* he    = fused + (long)BR * D;      // [BR][D]
  float* out   = (float*)d_out;             // [BR][D] == [B][R][D] flat

  dim3 grid(BR / 128, D / 64);  // 40 x 8 = 320 workgroups

  // Stage 1: fused = tanh([img|ques] @ W_fuse^T + b_fuse)
  gemm_tanh_kernel<true, false><<<grid, 256, 0, stream>>>(
      img, ques, W_fuse, b_fuse, nullptr, fused, BR, D, FUSION);

  // Stage 2: attention + weighted history sum
  attn_kernel<<<BR, 256, 0, stream>>>(hist, fused, w_att, he);

  // Stage 3: out = fused + tanh(he @ W_hist^T + b_hist)
  gemm_tanh_kernel<false, true><<<grid, 256, 0, stream>>>(
      he, nullptr, W_hist, b_hist, fused, out, BR, D, D);
}